// QuantumGate_65481071410733
// MI455X (gfx1250) — compile-verified
//
#include <hip/hip_runtime.h>
#include <hip/hip_bf16.h>
#include <math.h>

// ---------------------------------------------------------------------------
// 12-qubit statevector circuit, B=512.
// One workgroup (128 threads = 4 wave32) per batch element; state lives in LDS.
// 4-qubit gate groups -> 16x16 complex unitaries applied with
// V_WMMA_F32_16X16X4_F32 (4 real WMMAs per complex 16x16x16 K-chunk).
// CNOT chain per layer == Gray-code permutation: sigma(b) = b ^ (b>>1).
// Per-(layer,qubit) 2x2 unitaries are computed ONCE (48 of them) and the
// 16x16 Kronecker products are assembled from LDS, avoiding redundant
// transcendental evaluation.
// ---------------------------------------------------------------------------

typedef float v2f __attribute__((ext_vector_type(2)));
typedef float v8f __attribute__((ext_vector_type(8)));

#define NQ   12
#define DIM  4096      // 2^12 amplitudes
#define NL   4
#define NTHR 128       // 4 waves
#define MOFF 3072      // imag-plane offset inside gmat (12*256 floats)

// Map (group nibble j, remaining 8 bits r) -> full 12-bit basis index.
// group0 = qubits 0-3 = bits 11:8, group1 = bits 7:4, group2 = bits 3:0.
template <int GRP>
__device__ __forceinline__ int fmapc(int j, int r) {
  if constexpr (GRP == 0) return (j << 8) | r;
  else if constexpr (GRP == 1) return ((r & 0xF0) << 4) | (j << 4) | (r & 0xF);
  else return (r << 4) | j;
}

__device__ __forceinline__ v8f wmma4(v2f a, v2f b, v8f c) {
  // D(16x16,f32) = A(16x4,f32) * B(4x16,f32) + C
  return __builtin_amdgcn_wmma_f32_16x16x4_f32(
      /*neg_a=*/false, a, /*neg_b=*/false, b,
      /*c_mod=*/(short)0, c, /*reuse_a=*/false, /*reuse_b=*/false);
}

// Apply 16x16 complex unitary M to the state along group GRP's 4 bits.
// st: real plane at st[0..DIM), imag plane at st[DIM..2*DIM).
// m:  real 16x16 at m[0..256), imag at m[MOFF..MOFF+256).
// 16 column tiles of 16; tiles are disjoint across waves -> in-place is safe.
template <int GRP>
__device__ __forceinline__ void apply_group(float* __restrict__ st,
                                            const float* __restrict__ m,
                                            int lane, int wave) {
  const int row  = lane & 15;            // A-matrix row (M index)
  const int ksel = (lane >> 4) << 1;     // lanes 16-31 hold K+2 columns
  v2f ar4[4], ai4[4], an4[4];
#pragma unroll
  for (int c = 0; c < 4; ++c) {
    const int kb = row * 16 + 4 * c + ksel;
    v2f arr = { m[kb],        m[kb + 1] };
    v2f aii = { m[kb + MOFF], m[kb + 1 + MOFF] };
    ar4[c] = arr;
    ai4[c] = aii;
    an4[c] = -aii;                       // f32 WMMA has no A-negate bit
  }
#pragma unroll 2
  for (int tt = 0; tt < 4; ++tt) {
    const int col = (wave * 4 + tt) * 16 + (lane & 15);
    v8f dr = {};
    v8f di = {};
#pragma unroll
    for (int c = 0; c < 4; ++c) {
      const int kb = 4 * c + ksel;
      const int a0 = fmapc<GRP>(kb, col);
      const int a1 = fmapc<GRP>(kb + 1, col);
      v2f br = { st[a0],       st[a1] };
      v2f bi = { st[a0 + DIM], st[a1 + DIM] };
      dr = wmma4(ar4[c], br, dr);        // Dr += Mr*Sr
      dr = wmma4(an4[c], bi, dr);        // Dr += (-Mi)*Si
      di = wmma4(ar4[c], bi, di);        // Di += Mr*Si
      di = wmma4(ai4[c], br, di);        // Di += Mi*Sr
    }
#pragma unroll
    for (int v = 0; v < 8; ++v) {
      const int i = v + ((lane >> 4) << 3);
      const int addr = fmapc<GRP>(i, col);
      st[addr]       = dr[v];
      st[addr + DIM] = di[v];
    }
  }
}

__global__ __launch_bounds__(NTHR)
void qc12_kernel(const float* __restrict__ x,       // (B,12)
                 const float* __restrict__ params,  // (4,12,3)
                 const float* __restrict__ lnw,     // (1,)
                 const float* __restrict__ lnb,     // (1,)
                 float* __restrict__ out) {         // (B,)
  __shared__ float st[2 * DIM];        // 32 KB  state: real | imag
  __shared__ float gmat[2 * 12 * 256]; // 24 KB  matrices: real(12x256) | imag
  __shared__ float uq[48 * 8];         // 1.5 KB (layer,qubit) 2x2 unitaries
  __shared__ float cxs[NQ], sxs[NQ];
  __shared__ float red[NTHR];

  const int tid  = threadIdx.x;
  const int lane = tid & 31;
  const int wave = tid >> 5;
  const int bidx = blockIdx.x;

  // ---- per-qubit RX factors for this batch element --------------------------
  if (tid < NQ) {
    float h = 0.5f * x[bidx * NQ + tid];
    cxs[tid] = cosf(h);
    sxs[tid] = sinf(h);
  }

  // ---- 48 per-(layer,qubit) 2x2 unitaries U = Uz @ Uy -----------------------
  // u[0][0]=ez*cy  u[0][1]=-ez*sy  u[1][0]=conj(ez)*sy  u[1][1]=conj(ez)*cy
  if (tid < NL * NQ) {
    const int layer = tid / NQ, q = tid % NQ;
    const float ty = params[layer * 36 + q * 3 + 0] * 0.5f;
    const float tz = params[layer * 36 + q * 3 + 1] * 0.5f;
    const float cy = cosf(ty), sy = sinf(ty);
    const float ezr = cosf(tz), ezi = -sinf(tz);   // exp(-i tz)
    float* u = &uq[tid * 8];
    u[0] =  ezr * cy; u[1] =  ezi * cy;   // (0,0)
    u[2] = -ezr * sy; u[3] = -ezi * sy;   // (0,1)
    u[4] =  ezr * sy; u[5] = -ezi * sy;   // (1,0)
    u[6] =  ezr * cy; u[7] = -ezi * cy;   // (1,1)
  }
  __syncthreads();

  // ---- build 12 group unitaries: M = kron of 4 qubit gates ------------------
  for (int e = tid; e < 12 * 256; e += NTHR) {
    const int mat = e >> 8;            // layer*3 + grp
    const int layer = mat / 3, grp = mat % 3;
    const int idx = e & 255;
    const int i = idx >> 4, j = idx & 15;
    float mrv = 1.f, miv = 0.f;
#pragma unroll
    for (int t = 0; t < 4; ++t) {
      const int q = grp * 4 + t;
      const int bi_ = (i >> (3 - t)) & 1;
      const int bj  = (j >> (3 - t)) & 1;
      const float* u = &uq[((layer * NQ + q) * 4 + bi_ * 2 + bj) * 2];
      const float ur = u[0], ui = u[1];
      const float nr = mrv * ur - miv * ui;
      const float ni = mrv * ui + miv * ur;
      mrv = nr; miv = ni;
    }
    gmat[mat * 256 + idx]        = mrv;
    gmat[mat * 256 + idx + MOFF] = miv;
  }
  __syncthreads();

  // ---- initial product state: |psi> = kron_q RX(x_q)|0> ---------------------
  for (int b = tid; b < DIM; b += NTHR) {
    float ar = 1.f, ai = 0.f;
#pragma unroll
    for (int q = 0; q < NQ; ++q) {
      const int bit = (b >> (11 - q)) & 1;
      const float fr = bit ? 0.f : cxs[q];
      const float fi = bit ? -sxs[q] : 0.f;
      const float nr = ar * fr - ai * fi;
      const float ni = ar * fi + ai * fr;
      ar = nr; ai = ni;
    }
    st[b]       = ar;
    st[b + DIM] = ai;
  }
  __syncthreads();

  // ---- layers ---------------------------------------------------------------
  for (int l = 0; l < NL; ++l) {
    // Three commuting 16x16 complex unitaries (one per 4-qubit group).
    apply_group<0>(st, &gmat[(l * 3 + 0) * 256], lane, wave);
    __syncthreads();
    apply_group<1>(st, &gmat[(l * 3 + 1) * 256], lane, wave);
    __syncthreads();
    apply_group<2>(st, &gmat[(l * 3 + 2) * 256], lane, wave);
    __syncthreads();

    // CNOT chain q=0..10 composes to the Gray-code map: s_new[b] = s_old[b^(b>>1)]
    float pr[32], pim[32];
#pragma unroll
    for (int k = 0; k < 32; ++k) {
      const int b = k * NTHR + tid;
      const int s = b ^ (b >> 1);
      pr[k]  = st[s];
      pim[k] = st[s + DIM];
    }
    __syncthreads();
#pragma unroll
    for (int k = 0; k < 32; ++k) {
      const int b = k * NTHR + tid;
      st[b]       = pr[k];
      st[b + DIM] = pim[k];
    }
    __syncthreads();
  }

  // ---- <Z_0> = sum_{bit11=0}|a|^2 - sum_{bit11=1}|a|^2 ----------------------
  float acc = 0.f;
  for (int b = tid; b < DIM; b += NTHR) {
    const float v = st[b] * st[b] + st[b + DIM] * st[b + DIM];
    acc += (b & 2048) ? -v : v;
  }
  red[tid] = acc;
  __syncthreads();
  for (int s = NTHR / 2; s > 0; s >>= 1) {
    if (tid < s) red[tid] += red[tid + s];
    __syncthreads();
  }
  if (tid == 0) {
    const float ev = red[0];
    const float p = 1.f / (1.f + expf(-ev));
    // size-1 LayerNorm, computed faithfully (mu==p, var==0)
    const float mu = p;
    const float var = 0.f;
    out[bidx] = (p - mu) * rsqrtf(var + 1e-5f) * lnw[0] + lnb[0];
  }
}

extern "C" void kernel_launch(void* const* d_in, const int* in_sizes, int n_in,
                              void* d_out, int out_size, void* d_ws, size_t ws_size,
                              hipStream_t stream) {
  const float* x      = (const float*)d_in[0];   // (B,12) f32
  const float* params = (const float*)d_in[1];   // (4,12,3) f32
  const float* lnw    = (const float*)d_in[2];   // (1,) f32
  const float* lnb    = (const float*)d_in[3];   // (1,) f32
  float* out          = (float*)d_out;           // (B,) f32
  const int B = in_sizes[0] / NQ;
  qc12_kernel<<<B, NTHR, 0, stream>>>(x, params, lnw, lnb, out);
}